// WaveletPRVModule_16853451669784
// MI455X (gfx1250) — compile-verified
//
#include <hip/hip_runtime.h>
#include <math.h>

// WaveletPRV: per-row db4 level-1 detail coefficients (symmetric ext),
// affine, then max-min range per row.
// B=4096 rows, N=16384 -> NOUT=(N+7)/2=8195 detail coeffs per row.
//
// Memory-bound (268 MB @ 23.3 TB/s ~= 11.5 us floor, only 0.54 GFLOP), so we
// use the CDNA5 async tensor-data path: GLOBAL_LOAD_ASYNC_TO_LDS_B128 streams
// each row into LDS (no VGPR staging), non-temporal to avoid thrashing the
// 192 MB L2 with a 268 MB single-pass stream. Compute reads LDS with 4x reuse.

#define ROW_N   16384
#define NOUT    8195
#define TPB     256

__global__ __launch_bounds__(TPB) void wavelet_prv_kernel(
    const float* __restrict__ x,      // [B, ROW_N]
    const float* __restrict__ ww,     // [5,1] -> use [4]
    const float* __restrict__ wb,     // [5,1] -> use [4]
    float* __restrict__ out)          // [B]
{
    __shared__ float sh[ROW_N];       // exactly 64 KB
    const int tid = threadIdx.x;
    const int row = blockIdx.x;

    // ---- Async stream: global row -> LDS (ASYNCcnt path, 16B/lane/iter) ----
    uint32_t ldsbase = (uint32_t)(uintptr_t)(&sh[0]);   // flat LDS addr low 32b == LDS offset
    uint64_t gbase   = (uint64_t)(uintptr_t)(x + (size_t)row * ROW_N);
    uint32_t off     = (uint32_t)tid * 16u;
#pragma unroll
    for (int it = 0; it < 16; ++it) {
        uint32_t laddr = ldsbase + off;
        asm volatile("global_load_async_to_lds_b128 %0, %1, %2 th:TH_LOAD_NT"
                     :: "v"(laddr), "v"(off), "s"(gbase) : "memory");
        off += TPB * 16u;
    }
    asm volatile("s_wait_asynccnt 0x0" ::: "memory");
    __syncthreads();

    const float w = ww[4];
    const float b = wb[4];

    // db4 dec_hi taps: out[j] = sum_t h[t] * x_sym(2j+1-t)
    const float h0 = -0.23037781330885523f;
    const float h1 =  0.7148465705525415f;
    const float h2 = -0.6308807679295904f;
    const float h3 = -0.02798376941698385f;
    const float h4 =  0.18703481171888114f;
    const float h5 =  0.030841381835986965f;
    const float h6 = -0.032883011666982945f;
    const float h7 = -0.010597401784997278f;
    const float H[8] = {h0, h1, h2, h3, h4, h5, h6, h7};

    float mn =  INFINITY;
    float mx = -INFINITY;

    for (int j = tid; j < NOUT; j += TPB) {
        float acc;
        const int m0 = 2 * j + 1;                 // highest input index used
        if (j >= 4 && j <= NOUT - 5) {
            // interior: indices m0-7 .. m0 all within [0, ROW_N)
            const float* p = &sh[m0 - 7];
            acc = fmaf(h0, p[7],
                  fmaf(h1, p[6],
                  fmaf(h2, p[5],
                  fmaf(h3, p[4],
                  fmaf(h4, p[3],
                  fmaf(h5, p[2],
                  fmaf(h6, p[1], h7 * p[0])))))));
        } else {
            acc = 0.0f;
#pragma unroll
            for (int t = 0; t < 8; ++t) {
                int m = m0 - t;
                m = (m < 0) ? (-1 - m) : ((m >= ROW_N) ? (2 * ROW_N - 1 - m) : m);
                acc = fmaf(H[t], sh[m], acc);
            }
        }
        const float v = fmaf(acc, w, b);
        mn = fminf(mn, v);
        mx = fmaxf(mx, v);
    }

    // ---- wave32 reduction, then cross-wave through LDS ----
#pragma unroll
    for (int o = 16; o > 0; o >>= 1) {
        mn = fminf(mn, __shfl_down(mn, o, 32));
        mx = fmaxf(mx, __shfl_down(mx, o, 32));
    }
    __syncthreads();                 // all LDS row reads done; safe to reuse sh[]
    const int wave = tid >> 5;
    if ((tid & 31) == 0) {
        sh[2 * wave]     = mn;
        sh[2 * wave + 1] = mx;
    }
    __syncthreads();
    if (tid == 0) {
        float rmn = sh[0], rmx = sh[1];
#pragma unroll
        for (int i = 1; i < TPB / 32; ++i) {
            rmn = fminf(rmn, sh[2 * i]);
            rmx = fmaxf(rmx, sh[2 * i + 1]);
        }
        out[row] = rmx - rmn;
    }
}

extern "C" void kernel_launch(void* const* d_in, const int* in_sizes, int n_in,
                              void* d_out, int out_size, void* d_ws, size_t ws_size,
                              hipStream_t stream) {
    (void)n_in; (void)d_ws; (void)ws_size; (void)out_size;
    const float* x  = (const float*)d_in[0];
    const float* ww = (const float*)d_in[1];
    const float* wb = (const float*)d_in[2];
    float* out = (float*)d_out;
    const int rows = in_sizes[0] / ROW_N;   // 4096
    wavelet_prv_kernel<<<rows, TPB, 0, stream>>>(x, ww, wb, out);
}